// SentenceGraphProp_56160992362792
// MI455X (gfx1250) — compile-verified
//
#include <hip/hip_runtime.h>
#include <hip/hip_bf16.h>

typedef __attribute__((ext_vector_type(2))) float v2f;
typedef __attribute__((ext_vector_type(8))) float v8f;

#define D 128            // feature dim
#define BLK_ROWS 32      // rows (nodes) per GEMM block = 2 M-tiles of 16
#define A_PAD 130        // padded LDS row stride for A tile (bank-conflict break)

// ---------------------------------------------------------------------------
// Kernel 1: zero the accumulator (d_out doubles as the segment-sum buffer)
// ---------------------------------------------------------------------------
__global__ __launch_bounds__(256) void zero_kernel(float4* out, int n4) {
    int i = blockIdx.x * 256 + threadIdx.x;
    if (i < n4) out[i] = make_float4(0.f, 0.f, 0.f, 0.f);
}

// ---------------------------------------------------------------------------
// Kernel 2: gather x[col], scale by edge weight, scatter-add into acc[row].
// One wave (32 lanes) per edge; each lane handles 4 contiguous floats
// (float4 load = fully coalesced 512B row read, 4 hw f32 atomics).
// ---------------------------------------------------------------------------
__global__ __launch_bounds__(256) void scatter_kernel(const float* __restrict__ x,
                                                      const long long* __restrict__ ei,
                                                      const float* __restrict__ ew,
                                                      float* __restrict__ acc,
                                                      int E) {
    int t = blockIdx.x * 256 + threadIdx.x;
    int e = t >> 5;
    if (e >= E) return;
    int lane = t & 31;

    long long row = ei[e];       // destination node
    long long col = ei[E + e];   // source node
    float w = ew[e];

    const float4* src = (const float4*)(x + col * D);
    float4 v = src[lane];

    float* dst = acc + row * D + lane * 4;
    unsafeAtomicAdd(dst + 0, v.x * w);
    unsafeAtomicAdd(dst + 1, v.y * w);
    unsafeAtomicAdd(dst + 2, v.z * w);
    unsafeAtomicAdd(dst + 3, v.w * w);
}

// ---------------------------------------------------------------------------
// Kernel 3: in-place GEMM  io = io @ W^T + b  via V_WMMA_F32_16X16X4_F32.
// Each block owns BLK_ROWS rows: stage its A tile + transposed W into LDS,
// barrier, then each of the 8 waves computes one 16-col N-tile across both
// 16-row M-subtiles. In-place is safe: all reads of io happen before the
// barrier; each block writes only its own rows.
//
// Fragment layouts (ISA 7.12.2, 32-bit, wave32):
//   A 16x4 : lane L<16 -> M=L,   {K=k0,k0+1}; lane>=16 -> M=L-16, {k0+2,k0+3}
//   B 4x16 : lane L<16 -> N=L,   rows {k0,k0+1}; lane>=16 -> rows {k0+2,k0+3}
//   C/D    : VGPR v, lane L -> M = v + 8*(L>=16), N = L%16
// ---------------------------------------------------------------------------
__global__ __launch_bounds__(256) void gemm_wmma_kernel(const float* __restrict__ W,
                                                        const float* __restrict__ bias,
                                                        float* __restrict__ io) {
    __shared__ float sWt[D * D];            // W transposed: sWt[k*128 + n] = W[n*128 + k]
    __shared__ float sA[BLK_ROWS * A_PAD];  // A tile, padded rows

    const int tid  = threadIdx.x;
    const int wave = tid >> 5;       // 0..7 -> N-tile
    const int lane = tid & 31;
    const long long rowbase = (long long)blockIdx.x * BLK_ROWS;

    // Stage W transposed: coalesced LDS writes, strided (cached) global reads.
    for (int i = tid; i < D * D; i += 256) {
        int k = i >> 7, n = i & 127;
        sWt[i] = W[n * D + k];
    }
    // Stage this block's A tile (reads complete before any output store).
    for (int i = tid; i < BLK_ROWS * D; i += 256) {
        int r = i >> 7, c = i & 127;
        sA[r * A_PAD + c] = io[(rowbase + r) * D + c];
    }
    __syncthreads();

    const int n0   = wave * 16;
    const int half = lane >> 4;      // 0: K pair {k0,k0+1}, 1: {k0+2,k0+3}
    const int l16  = lane & 15;
    const float bv = bias[n0 + l16];

    #pragma unroll
    for (int mt = 0; mt < BLK_ROWS / 16; ++mt) {
        v8f c = {};
        const int arow = mt * 16 + l16;

        #pragma unroll
        for (int k0 = 0; k0 < D; k0 += 4) {
            const int kb = k0 + half * 2;
            v2f a, b;
            a.x = sA[arow * A_PAD + kb];
            a.y = sA[arow * A_PAD + kb + 1];
            b.x = sWt[kb * D + n0 + l16];
            b.y = sWt[(kb + 1) * D + n0 + l16];
            c = __builtin_amdgcn_wmma_f32_16x16x4_f32(
                    false, a, false, b, (short)0, c, false, false);
        }

        const long long mb = rowbase + mt * 16 + half * 8;
        #pragma unroll
        for (int v = 0; v < 8; ++v) {
            io[(mb + v) * D + n0 + l16] = c[v] + bv;
        }
    }
}

// ---------------------------------------------------------------------------
extern "C" void kernel_launch(void* const* d_in, const int* in_sizes, int n_in,
                              void* d_out, int out_size, void* d_ws, size_t ws_size,
                              hipStream_t stream) {
    const float*      x  = (const float*)d_in[0];
    const long long*  ei = (const long long*)d_in[1];
    const float*      ew = (const float*)d_in[2];
    const float*      W  = (const float*)d_in[3];
    const float*      b  = (const float*)d_in[4];
    float*            out = (float*)d_out;

    const int N = in_sizes[0] / D;   // 100000 nodes
    const int E = in_sizes[2];       // 600000 edges

    // 1) zero accumulator (== d_out)
    const int n4 = (N * D) / 4;
    zero_kernel<<<(n4 + 255) / 256, 256, 0, stream>>>((float4*)out, n4);

    // 2) scatter-add weighted messages (one wave per edge)
    const long long threads = (long long)E * 32;
    scatter_kernel<<<(unsigned)((threads + 255) / 256), 256, 0, stream>>>(x, ei, ew, out, E);

    // 3) in-place WMMA GEMM + bias
    gemm_wmma_kernel<<<N / BLK_ROWS, 256, 0, stream>>>(W, b, out);
}